// sru_53214644798204
// MI455X (gfx1250) — compile-verified
//
#include <hip/hip_runtime.h>
#include <hip/hip_bf16.h>

// ---------------------------------------------------------------------------
// SRU single-query attention + FFN, restructured for MI455X (gfx1250).
//
// Key restructure (mathematically exact):
//   scores[b,h,l] = SCALE * <qk[b,h,:], y[b,:,l]>,  qk = (x Wq^T + bq)_h^T Wk_h
//   (the q.bk term is constant over l -> softmax invariant -> dropped)
//   att[b,h,:]    = Wv_h @ u[b,h,:] + bv_h,  u[b,h,c] = sum_l alpha[b,h,l] y[b,c,l]
// This removes the 275-GFLOP fp32 K/V GEMMs; the job becomes two streaming
// passes over y (2 x 256 MiB -> ~22 us at 23.3 TB/s). Both passes use
// V_WMMA_F32_16X16X4_F32 (M=16 == NHEAD) so the 8 FLOP/byte inner product
// runs on the matrix pipe at full fp32 precision, not the VALU.
//
// Workspace requirement: ~13.4 MB.
// ---------------------------------------------------------------------------

typedef float v2f __attribute__((ext_vector_type(2)));
typedef float v8f __attribute__((ext_vector_type(8)));

#define B_     32
#define NHID   1024
#define LSEQ   2048
#define NHEAD  16
#define DIN    4096
#define SCALE  0.125f                  // 1/sqrt(64)
#define INV_BN 0.9999950000374997f     // 1/sqrt(1+1e-5)

// ---------------------------------------------------------------------------
// Generic small GEMM: out[b,o] = epilogue( sum_c X[b,c]*W[o,c] + bias[o] )
// B=32 rows, 32 o-columns per block, C chunked through LDS.
// mode 0: plain (q projection), 1: relu (FFN1), 2: (+bw +ret)*g*INV_BN+b (FFN2)
// ---------------------------------------------------------------------------
__global__ void __launch_bounds__(256)
gemm_bt(float* __restrict__ out, const float* __restrict__ X,
        const float* __restrict__ W, const float* __restrict__ bias,
        int C, int O, int mode,
        const float* __restrict__ resid, const float* __restrict__ gbn,
        const float* __restrict__ bbn)
{
    const int o_block = blockIdx.x * 32;
    const int t  = threadIdx.x;
    const int b  = t & 31;
    const int og = t >> 5;

    __shared__ float x_s[128 * 33];   // X chunk transposed: [c][b], pitch 33
    __shared__ float w_s[32 * 132];   // W chunk: [o_local][c], pitch 132

    float acc[4] = {0.f, 0.f, 0.f, 0.f};

    for (int c0 = 0; c0 < C; c0 += 128) {
        {   // stage X (32 b x 128 c), coalesced loads, transposed store
            const int br = t >> 3, cb = (t & 7) * 16;
            const float* src = X + (size_t)br * C + c0 + cb;
#pragma unroll
            for (int j = 0; j < 16; ++j) x_s[(cb + j) * 33 + br] = src[j];
        }
        {   // stage W (32 o x 128 c)
            const int ol = t >> 3, cb = (t & 7) * 16;
            const float* src = W + (size_t)(o_block + ol) * C + c0 + cb;
#pragma unroll
            for (int j = 0; j < 16; ++j) w_s[ol * 132 + cb + j] = src[j];
        }
        __syncthreads();
        for (int c = 0; c < 128; ++c) {
            const float xv = x_s[c * 33 + b];
#pragma unroll
            for (int mi = 0; mi < 4; ++mi)
                acc[mi] += xv * w_s[(og + 8 * mi) * 132 + c];
        }
        __syncthreads();
    }

#pragma unroll
    for (int mi = 0; mi < 4; ++mi) {
        const int o = o_block + og + 8 * mi;
        float v = acc[mi] + bias[o];
        if (mode == 1) {
            v = fmaxf(v, 0.f);
        } else if (mode == 2) {
            v = (v + resid[(size_t)b * O + o]) * (gbn[o] * INV_BN) + bbn[o];
        }
        out[(size_t)b * O + o] = v;
    }
}

// ---------------------------------------------------------------------------
// qk[b,h,c] = sum_d q[b,h*64+d] * Wk[h*64+d, c]   (contraction over Wk rows)
// grid (NHEAD, NHID/256); thread = one c; 32 batch accumulators in registers.
// ---------------------------------------------------------------------------
__global__ void __launch_bounds__(256)
qk_proj(const float* __restrict__ q, const float* __restrict__ Wk,
        float* __restrict__ qk)
{
    const int h  = blockIdx.x;
    const int c0 = blockIdx.y * 256;
    const int t  = threadIdx.x;

    __shared__ float q_s[32 * 64];
    {
        const int br = t >> 3, db = (t & 7) * 8;
        const float* src = q + (size_t)br * NHID + h * 64 + db;
#pragma unroll
        for (int j = 0; j < 8; ++j) q_s[br * 64 + db + j] = src[j];
    }
    __syncthreads();

    float acc[32];
#pragma unroll
    for (int b = 0; b < 32; ++b) acc[b] = 0.f;

    const float* wp = Wk + (size_t)(h * 64) * NHID + c0 + t;
    for (int d = 0; d < 64; ++d) {
        const float wv = wp[(size_t)d * NHID];
#pragma unroll
        for (int b = 0; b < 32; ++b) acc[b] += q_s[b * 64 + d] * wv;
    }
#pragma unroll
    for (int b = 0; b < 32; ++b)
        qk[(size_t)(b * NHEAD + h) * NHID + c0 + t] = acc[b];
}

// ---------------------------------------------------------------------------
// Pass 1 (WMMA, streams y once):
//   scores[b,h,l] = SCALE * sum_c qk[b,h,c] * y[b,c,l]
// grid (B, LSEQ/128); 8 waves x one 16(H)x16(L) tile each; K chunked by 256
// with qk staged transposed in LDS ([c][h], pitch 17 -> conflict-free A frags).
// B fragments read directly from global y (lanes 0-15 contiguous in l).
// ---------------------------------------------------------------------------
__global__ void __launch_bounds__(256)
pass1_scores(const float* __restrict__ y, const float* __restrict__ qk,
             float* __restrict__ scores)
{
    const int b    = blockIdx.x;
    const int l0   = blockIdx.y * 128;
    const int t    = threadIdx.x;
    const int wave = t >> 5;
    const int lane = t & 31;
    const int nl   = lane & 15;          // M (head) for A-frag, N (l) for B/D
    const int kl2  = (lane >> 4) * 2;    // K sub-offset: 0 or 2

    __shared__ float qk_s[256 * 17];     // [c_local][h]

    const float* yrow = y + (size_t)b * NHID * LSEQ + l0 + wave * 16 + nl;

    v8f acc = {};

    for (int c0 = 0; c0 < NHID; c0 += 256) {
        {   // stage qk chunk (16 h x 256 c), coalesced row loads
            const int hh = t >> 4, cb = (t & 15) * 16;
            const float* src = qk + (size_t)(b * NHEAD + hh) * NHID + c0 + cb;
#pragma unroll
            for (int j = 0; j < 16; ++j) qk_s[(cb + j) * 17 + hh] = src[j];
        }
        __syncthreads();

        const float* yc = yrow + (size_t)(c0 + kl2) * LSEQ;
#pragma unroll 4
        for (int kk = 0; kk < 256; kk += 4) {
            const int ck = kk + kl2;
            v2f a, bb;
            a[0]  = qk_s[ck * 17 + nl];
            a[1]  = qk_s[(ck + 1) * 17 + nl];
            bb[0] = yc[(size_t)kk * LSEQ];
            bb[1] = yc[(size_t)kk * LSEQ + LSEQ];
            acc = __builtin_amdgcn_wmma_f32_16x16x4_f32(
                false, a, false, bb, (short)0, acc, false, false);
        }
        __syncthreads();
    }

    const int mbase = (lane >> 4) * 8;
    float* sp = scores + (size_t)(b * NHEAD + mbase) * LSEQ + l0 + wave * 16 + nl;
#pragma unroll
    for (int r = 0; r < 8; ++r) sp[(size_t)r * LSEQ] = acc[r] * SCALE;
}

// ---------------------------------------------------------------------------
// Row-wise softmax over L=2048; one block per (b,h) row.
// ---------------------------------------------------------------------------
__global__ void __launch_bounds__(256)
softmax_rows(const float* __restrict__ s, float* __restrict__ a)
{
    const int row = blockIdx.x;
    const int t   = threadIdx.x;
    const float* sp = s + (size_t)row * LSEQ;
    float* ap = a + (size_t)row * LSEQ;

    __shared__ float red[256];

    float m = -1e30f;
    for (int i = t; i < LSEQ; i += 256) m = fmaxf(m, sp[i]);
    red[t] = m; __syncthreads();
    for (int o = 128; o; o >>= 1) {
        if (t < o) red[t] = fmaxf(red[t], red[t + o]);
        __syncthreads();
    }
    m = red[0];
    __syncthreads();

    float sum = 0.f;
    for (int i = t; i < LSEQ; i += 256) {
        const float e = __expf(sp[i] - m);
        ap[i] = e;
        sum += e;
    }
    red[t] = sum; __syncthreads();
    for (int o = 128; o; o >>= 1) {
        if (t < o) red[t] += red[t + o];
        __syncthreads();
    }
    const float inv = 1.f / red[0];
    for (int i = t; i < LSEQ; i += 256) ap[i] *= inv;
}

// ---------------------------------------------------------------------------
// Pass 2 (WMMA, streams y again):
//   u[b,h,c] = sum_l alphas[b,h,l] * y[b,c,l]     (M=h, N=c, K=l)
// grid (B, NHID/128); L chunked by 64: y tile (128c x 64l) staged transposed
// into LDS ([l][c], pitch 129, coalesced b128 global loads), alphas chunk
// staged as [l][h] pitch 17.
// ---------------------------------------------------------------------------
__global__ void __launch_bounds__(256)
pass2_u(const float* __restrict__ y, const float* __restrict__ alphas,
        float* __restrict__ u)
{
    const int b    = blockIdx.x;
    const int c0   = blockIdx.y * 128;
    const int t    = threadIdx.x;
    const int wave = t >> 5;
    const int lane = t & 31;
    const int nl   = lane & 15;
    const int kl2  = (lane >> 4) * 2;

    __shared__ float y_s[64 * 129];   // [l_local][c_local]
    __shared__ float al_s[64 * 17];   // [l_local][h]

    v8f acc = {};

    for (int l0 = 0; l0 < LSEQ; l0 += 64) {
        {   // stage alphas chunk (16 h x 64 l)
            const int hh = t >> 4, lb = (t & 15) * 4;
            const float* src = alphas + (size_t)(b * NHEAD + hh) * LSEQ + l0 + lb;
#pragma unroll
            for (int j = 0; j < 4; ++j) al_s[(lb + j) * 17 + hh] = src[j];
        }
        {   // stage y tile (128 c x 64 l), transpose into LDS
            const int cr = t >> 1, lb = (t & 1) * 32;
            const float* src = y + (size_t)(b * NHID + c0 + cr) * LSEQ + l0 + lb;
#pragma unroll
            for (int j = 0; j < 32; ++j) y_s[(lb + j) * 129 + cr] = src[j];
        }
        __syncthreads();

#pragma unroll 4
        for (int kk = 0; kk < 64; kk += 4) {
            const int k2 = kk + kl2;
            v2f a, bb;
            a[0]  = al_s[k2 * 17 + nl];
            a[1]  = al_s[(k2 + 1) * 17 + nl];
            bb[0] = y_s[k2 * 129 + wave * 16 + nl];
            bb[1] = y_s[(k2 + 1) * 129 + wave * 16 + nl];
            acc = __builtin_amdgcn_wmma_f32_16x16x4_f32(
                false, a, false, bb, (short)0, acc, false, false);
        }
        __syncthreads();
    }

    const int mbase = (lane >> 4) * 8;
    float* up = u + (size_t)(b * NHEAD + mbase) * NHID + c0 + wave * 16 + nl;
#pragma unroll
    for (int r = 0; r < 8; ++r) up[(size_t)r * NHID] = acc[r];
}

// ---------------------------------------------------------------------------
// ret[b,o] = (x[b,o] + Wv[o,:].u[b,h,:] + bv[o]) * (g1[o]*INV_BN) + b1[o]
// one wave per output channel o; Wv row held in registers, reused over 32 b.
// ---------------------------------------------------------------------------
__global__ void __launch_bounds__(256)
attn_out(const float* __restrict__ u, const float* __restrict__ Wv,
         const float* __restrict__ bv, const float* __restrict__ x,
         const float* __restrict__ g1, const float* __restrict__ b1,
         float* __restrict__ ret)
{
    const int wave = threadIdx.x >> 5;
    const int lane = threadIdx.x & 31;
    const int o = blockIdx.x * 8 + wave;      // 0..1023
    const int h = o >> 6;

    const float4* wr = (const float4*)(Wv + (size_t)o * NHID);
    float4 w[8];
#pragma unroll
    for (int j = 0; j < 8; ++j) w[j] = wr[lane + 32 * j];

    const float gi  = g1[o] * INV_BN;
    const float bb  = b1[o];
    const float bvo = bv[o];

    for (int b = 0; b < 32; ++b) {
        const float4* up = (const float4*)(u + (size_t)(b * NHEAD + h) * NHID);
        float p = 0.f;
#pragma unroll
        for (int j = 0; j < 8; ++j) {
            const float4 uv = up[lane + 32 * j];
            p += w[j].x * uv.x + w[j].y * uv.y + w[j].z * uv.z + w[j].w * uv.w;
        }
#pragma unroll
        for (int off = 16; off; off >>= 1) p += __shfl_down(p, off, 32);
        if (lane == 0)
            ret[(size_t)b * NHID + o] = (x[(size_t)b * NHID + o] + p + bvo) * gi + bb;
    }
}

// ---------------------------------------------------------------------------
extern "C" void kernel_launch(void* const* d_in, const int* in_sizes, int n_in,
                              void* d_out, int out_size, void* d_ws, size_t ws_size,
                              hipStream_t stream)
{
    (void)in_sizes; (void)n_in; (void)out_size; (void)ws_size;

    const float* x   = (const float*)d_in[0];
    const float* y   = (const float*)d_in[1];
    const float* Wq  = (const float*)d_in[2];
    const float* bq  = (const float*)d_in[3];
    const float* Wk  = (const float*)d_in[4];
    // d_in[5] = bk: constant over l inside softmax -> exactly cancels, unused.
    const float* Wv  = (const float*)d_in[6];
    const float* bv  = (const float*)d_in[7];
    const float* g1  = (const float*)d_in[8];
    const float* b1  = (const float*)d_in[9];
    const float* w1  = (const float*)d_in[10];
    const float* bw1 = (const float*)d_in[11];
    const float* w2  = (const float*)d_in[12];
    const float* bw2 = (const float*)d_in[13];
    const float* g2  = (const float*)d_in[14];
    const float* b2  = (const float*)d_in[15];
    float* out = (float*)d_out;

    // workspace carve-up (floats); total ~13.4 MB
    float* ws     = (float*)d_ws;
    float* q      = ws;                    // 32 x 1024
    float* qk     = q      + 32768;        // 32 x 16 x 1024
    float* scores = qk     + 524288;       // 32 x 16 x 2048
    float* alphas = scores + 1048576;      // 32 x 16 x 2048
    float* u      = alphas + 1048576;      // 32 x 16 x 1024
    float* ret    = u      + 524288;       // 32 x 1024
    float* hbuf   = ret    + 32768;        // 32 x 4096

    // 1) q = x @ Wq^T + bq
    gemm_bt<<<dim3(NHID / 32), 256, 0, stream>>>(
        q, x, Wq, bq, NHID, NHID, 0, nullptr, nullptr, nullptr);

    // 2) qk[b,h,:] = q[b,h,:]^T Wk_h
    qk_proj<<<dim3(NHEAD, NHID / 256), 256, 0, stream>>>(q, Wk, qk);

    // 3) scores = SCALE * qk . y        (WMMA pass 1, streams y)
    pass1_scores<<<dim3(B_, LSEQ / 128), 256, 0, stream>>>(y, qk, scores);

    // 4) alphas = softmax(scores) over L
    softmax_rows<<<dim3(B_ * NHEAD), 256, 0, stream>>>(scores, alphas);

    // 5) u = y . alphas                 (WMMA pass 2, streams y)
    pass2_u<<<dim3(B_, NHID / 128), 256, 0, stream>>>(y, alphas, u);

    // 6) ret = (x + Wv.u + bv) * g1*INV_BN + b1
    attn_out<<<dim3(NHID / 8), 256, 0, stream>>>(u, Wv, bv, x, g1, b1, ret);

    // 7) h = relu(ret @ w1^T + bw1)
    gemm_bt<<<dim3(DIN / 32), 256, 0, stream>>>(
        hbuf, ret, w1, bw1, NHID, DIN, 1, nullptr, nullptr, nullptr);

    // 8) out = ((h @ w2^T + bw2) + ret) * g2*INV_BN + b2
    gemm_bt<<<dim3(NHID / 32), 256, 0, stream>>>(
        out, hbuf, w2, bw2, DIN, NHID, 2, ret, g2, b2);
}